// CNF_14456859919045
// MI455X (gfx1250) — compile-verified
//
#include <hip/hip_runtime.h>
#include <math.h>

typedef __attribute__((ext_vector_type(2))) float v2f;
typedef __attribute__((ext_vector_type(8))) float v8f;

#define DT (-0.1f)
#define LOG_2PI 1.8378770664093453f

// Tsit5 tableau
__constant__ float c_TA[6][5] = {
  {0.f, 0.f, 0.f, 0.f, 0.f},
  {0.161f, 0.f, 0.f, 0.f, 0.f},
  {-0.008480655492356989f, 0.335480655492357f, 0.f, 0.f, 0.f},
  {2.8971530571054935f, -6.359448489975075f, 4.3622954328695815f, 0.f, 0.f},
  {5.325864828439257f, -11.748883564062828f, 7.4955393428898365f, -0.09249506636175525f, 0.f},
  {5.86145544294642f, -12.92096931784711f, 8.159367898576159f, -0.071584973281401f, -0.028269050394068383f},
};
__constant__ float c_TC[6] = {0.0f, 0.161f, 0.327f, 0.9f, 0.9800255409045097f, 1.0f};
__constant__ float c_TB[6] = {0.09646076681806523f, 0.01f, 0.4798896504144996f,
                              1.379008574103742f, -3.290069515436081f, 2.324710524099774f};

// ---- LDS layout (float indices); padded pitches keep ds_load_b64 frags conflict-light ----
#define OFF_W1   0         // W1_1 (128x64), pitch 66   -> 8448
#define OFF_W2   8448      // W1_2 (128x128), pitch 130 -> 16640
#define OFF_W3   25088     // W1_3 (32x128), pitch 130  -> 4160
#define OFF_BT   29248     // W1_1[:, :32]^T (32x128), pitch 130 -> 4160
#define OFF_HIN  33408     // 64  (zi ++ cond)
#define OFF_H1   33472     // 128
#define OFF_H2   33600     // 128
#define OFF_G1   33728     // 128
#define OFF_WT1  33856     // 128
#define OFF_S1   33984     // 128 (8B-aligned base for v2f loads)
#define OFF_G2   34112     // 128
#define OFF_WT2  34240     // 128
#define OFF_S2   34368     // 128
#define OFF_G3   34496     // 32
#define OFF_WT3  34528     // 32
#define OFF_B1A  34560     // 128
#define OFF_W2CA 34688     // 128
#define OFF_B2A  34816     // 128
#define OFF_W3CA 34944     // 128
#define OFF_B1B  35072     // 128
#define OFF_W2CB 35200     // 128
#define OFF_B2B  35328     // 128
#define OFF_W3CB 35456     // 128
#define OFF_B1C  35584     // 32
#define OFF_W2CC 35616     // 32
#define OFF_B2C  35648     // 32
#define OFF_W3CC 35680     // 32
#define OFF_Z    35712     // 32
#define OFF_KZ   35744     // 6*32
#define OFF_KL   35936     // 6
#define OFF_RED  35942     // 128
#define SMEM_FLOATS 36070
#define SMEM_BYTES  (SMEM_FLOATS * sizeof(float))

__device__ __forceinline__ float sigmoidf_(float x) { return 1.0f / (1.0f + expf(-x)); }

__global__ __launch_bounds__(128)
void cnf_logp_kernel(const float* __restrict__ gZ, const float* __restrict__ gCond,
                     const float* __restrict__ W1a, const float* __restrict__ b1a,
                     const float* __restrict__ w2a, const float* __restrict__ b2a,
                     const float* __restrict__ w3a,
                     const float* __restrict__ W1b, const float* __restrict__ b1b,
                     const float* __restrict__ w2b, const float* __restrict__ b2b,
                     const float* __restrict__ w3b,
                     const float* __restrict__ W1c, const float* __restrict__ b1c,
                     const float* __restrict__ w2c, const float* __restrict__ b2c,
                     const float* __restrict__ w3c,
                     float* __restrict__ out)
{
  extern __shared__ float sm[];
  float* sW1  = sm + OFF_W1;   float* sW2  = sm + OFF_W2;
  float* sW3  = sm + OFF_W3;   float* sBT  = sm + OFF_BT;
  float* sHin = sm + OFF_HIN;  float* sH1  = sm + OFF_H1;   float* sH2 = sm + OFF_H2;
  float* sG1  = sm + OFF_G1;   float* sWt1 = sm + OFF_WT1;  float* sS1 = sm + OFF_S1;
  float* sG2  = sm + OFF_G2;   float* sWt2 = sm + OFF_WT2;  float* sS2 = sm + OFF_S2;
  float* sG3  = sm + OFF_G3;   float* sWt3 = sm + OFF_WT3;
  float* sB1a = sm + OFF_B1A;  float* sW2ca= sm + OFF_W2CA; float* sB2a= sm + OFF_B2A; float* sW3ca = sm + OFF_W3CA;
  float* sB1b = sm + OFF_B1B;  float* sW2cb= sm + OFF_W2CB; float* sB2b= sm + OFF_B2B; float* sW3cb = sm + OFF_W3CB;
  float* sB1c = sm + OFF_B1C;  float* sW2cc= sm + OFF_W2CC; float* sB2c= sm + OFF_B2C; float* sW3cc = sm + OFF_W3CC;
  float* sZ   = sm + OFF_Z;    float* sKz  = sm + OFF_KZ;   float* sKl = sm + OFF_KL;  float* sRed  = sm + OFF_RED;

  const int tid  = threadIdx.x;
  const int bidx = blockIdx.x;
  const int wave = tid >> 5;
  const int lane = tid & 31;
  const int lr   = lane & 15;
  const int half = lane >> 4;

  // ---- stage weights into LDS (padded pitches) ----
  for (int idx = tid; idx < 128 * 64;  idx += 128) sW1[(idx >> 6) * 66  + (idx & 63)]  = W1a[idx];
  for (int idx = tid; idx < 128 * 128; idx += 128) sW2[(idx >> 7) * 130 + (idx & 127)] = W1b[idx];
  for (int idx = tid; idx < 32 * 128;  idx += 128) sW3[(idx >> 7) * 130 + (idx & 127)] = W1c[idx];
  for (int idx = tid; idx < 32 * 128;  idx += 128) {
    int n = idx >> 7, k = idx & 127;
    sBT[n * 130 + k] = W1a[k * 64 + n];      // B-matrix = W1_1[:, :32] transposed
  }
  sB1a[tid] = b1a[tid]; sW2ca[tid] = w2a[tid]; sB2a[tid] = b2a[tid]; sW3ca[tid] = w3a[tid];
  sB1b[tid] = b1b[tid]; sW2cb[tid] = w2b[tid]; sB2b[tid] = b2b[tid]; sW3cb[tid] = w3b[tid];
  if (tid < 32) {
    sB1c[tid] = b1c[tid]; sW2cc[tid] = w2c[tid]; sB2c[tid] = b2c[tid]; sW3cc[tid] = w3c[tid];
    sZ[tid]       = gZ[bidx * 32 + tid];
    sHin[32 + tid]= gCond[bidx * 32 + tid];
  }
  __syncthreads();

  float logp = 0.0f;

  for (int s = 0; s < 10; ++s) {
    const float t0 = 1.0f - 0.1f * (float)s;
    for (int st = 0; st < 6; ++st) {
      // --- stage input zi ---
      if (tid < 32) {
        float v = sZ[tid];
        for (int j = 0; j < st; ++j) v += (DT * c_TA[st][j]) * sKz[j * 32 + tid];
        sHin[tid] = v;
      }
      __syncthreads();

      const float ti = t0 + c_TC[st] * DT;

      // --- t-dependent gates (per output row) ---
      {
        sG1[tid]  = sigmoidf_(sW2ca[tid] * ti + sB2a[tid]);
        sWt1[tid] = sW3ca[tid] * ti;
        sG2[tid]  = sigmoidf_(sW2cb[tid] * ti + sB2b[tid]);
        sWt2[tid] = sW3cb[tid] * ti;
        if (tid < 32) {
          sG3[tid]  = sigmoidf_(sW2cc[tid] * ti + sB2c[tid]);
          sWt3[tid] = sW3cc[tid] * ti;
        }
      }
      __syncthreads();

      // --- forward layer 1 ---
      {
        float u = sB1a[tid];
        #pragma unroll 8
        for (int c = 0; c < 64; ++c) u += sW1[tid * 66 + c] * sHin[c];
        float h = tanhf(u * sG1[tid] + sWt1[tid]);
        sH1[tid] = h;
        sS1[tid] = (1.0f - h * h) * sG1[tid];
      }
      __syncthreads();

      // --- forward layer 2 ---
      {
        float u = sB1b[tid];
        #pragma unroll 8
        for (int k = 0; k < 128; ++k) u += sW2[tid * 130 + k] * sH1[k];
        float h = tanhf(u * sG2[tid] + sWt2[tid]);
        sH2[tid] = h;
        sS2[tid] = (1.0f - h * h) * sG2[tid];
      }
      __syncthreads();

      // --- forward layer 3 -> kz ---
      if (tid < 32) {
        float u = sB1c[tid];
        #pragma unroll 8
        for (int k = 0; k < 128; ++k) u += sW3[tid * 130 + k] * sH2[k];
        sKz[st * 32 + tid] = u * sG3[tid] + sWt3[tid];
      }

      // --- Jacobian matmul: P = (W1_2 * diag(s1)) @ W1_1[:,:32]  via f32 WMMA ---
      // wave w owns output rows [32w, 32w+31], all 32 columns: 2x2 tiles of 16x16.
      v8f acc00 = {0.f,0.f,0.f,0.f,0.f,0.f,0.f,0.f};
      v8f acc01 = {0.f,0.f,0.f,0.f,0.f,0.f,0.f,0.f};
      v8f acc10 = {0.f,0.f,0.f,0.f,0.f,0.f,0.f,0.f};
      v8f acc11 = {0.f,0.f,0.f,0.f,0.f,0.f,0.f,0.f};
      const int mrow0 = wave * 32 + lr;
      const int mrow1 = mrow0 + 16;
      const int ncol0 = lr;
      const int ncol1 = lr + 16;
      #pragma unroll 4
      for (int k0 = 0; k0 < 128; k0 += 4) {
        const int k = k0 + half * 2;                    // A/B 16x16x4 K-slot per lane-half
        v2f sv = *(const v2f*)&sS1[k];                  // fold diag(s1) into A
        v2f a0 = *(const v2f*)&sW2[mrow0 * 130 + k]; a0 *= sv;
        v2f a1 = *(const v2f*)&sW2[mrow1 * 130 + k]; a1 *= sv;
        v2f b0 = *(const v2f*)&sBT[ncol0 * 130 + k];
        v2f b1 = *(const v2f*)&sBT[ncol1 * 130 + k];
        acc00 = __builtin_amdgcn_wmma_f32_16x16x4_f32(false, a0, false, b0, (short)0, acc00, false, false);
        acc01 = __builtin_amdgcn_wmma_f32_16x16x4_f32(false, a0, false, b1, (short)0, acc01, false, false);
        acc10 = __builtin_amdgcn_wmma_f32_16x16x4_f32(false, a1, false, b0, (short)0, acc10, false, false);
        acc11 = __builtin_amdgcn_wmma_f32_16x16x4_f32(false, a1, false, b1, (short)0, acc11, false, false);
      }

      // --- trace = sum g3[i] * W1_3[i,k] * s2[k] * P[k,i], contracted from fragments ---
      float part = 0.0f;
      {
        const float g0 = sG3[ncol0];
        const float g1 = sG3[ncol1];
        #pragma unroll
        for (int r = 0; r < 8; ++r) {
          const int kA = wave * 32 + half * 8 + r;      // rows of M-tile 0
          const int kB = kA + 16;                       // rows of M-tile 1
          const float sA = sS2[kA];
          const float sB = sS2[kB];
          part += acc00[r] * sA * g0 * sW3[ncol0 * 130 + kA];
          part += acc01[r] * sA * g1 * sW3[ncol1 * 130 + kA];
          part += acc10[r] * sB * g0 * sW3[ncol0 * 130 + kB];
          part += acc11[r] * sB * g1 * sW3[ncol1 * 130 + kB];
        }
      }
      sRed[tid] = part;
      __syncthreads();
      if (tid < 32) sRed[tid] = sRed[tid] + sRed[tid + 32] + sRed[tid + 64] + sRed[tid + 96];
      __syncthreads();
      if (tid == 0) {
        float tr = 0.0f;
        for (int q = 0; q < 32; ++q) tr += sRed[q];
        sKl[st] = tr;
      }
      __syncthreads();
    } // stages

    // --- Tsit5 solution update ---
    if (tid < 32) {
      float v = sZ[tid];
      #pragma unroll
      for (int i = 0; i < 6; ++i) v += (DT * c_TB[i]) * sKz[i * 32 + tid];
      sZ[tid] = v;
    }
    float dl = 0.0f;
    #pragma unroll
    for (int i = 0; i < 6; ++i) dl += c_TB[i] * sKl[i];
    logp += DT * dl;
    __syncthreads();
  } // steps

  // --- base log-prob of standard normal at z_final ---
  if (tid < 32) {
    const float zf = sZ[tid];
    sRed[tid] = -0.5f * zf * zf;
  }
  __syncthreads();
  if (tid == 0) {
    float blp = 0.0f;
    for (int q = 0; q < 32; ++q) blp += sRed[q];
    blp += 32.0f * (-0.5f * LOG_2PI);
    out[bidx] = logp + blp;
  }
}

extern "C" void kernel_launch(void* const* d_in, const int* in_sizes, int n_in,
                              void* d_out, int out_size, void* d_ws, size_t ws_size,
                              hipStream_t stream) {
  (void)in_sizes; (void)n_in; (void)out_size; (void)d_ws; (void)ws_size;
  const float* z    = (const float*)d_in[0];
  const float* cond = (const float*)d_in[1];
  const float* W1a  = (const float*)d_in[2];   // (128,64)
  const float* b1a  = (const float*)d_in[3];   // (128,)
  const float* w2a  = (const float*)d_in[4];   // (128,1)
  const float* b2a  = (const float*)d_in[5];   // (128,)
  const float* w3a  = (const float*)d_in[6];   // (128,1)
  const float* W1b  = (const float*)d_in[7];   // (128,128)
  const float* b1b  = (const float*)d_in[8];
  const float* w2b  = (const float*)d_in[9];
  const float* b2b  = (const float*)d_in[10];
  const float* w3b  = (const float*)d_in[11];
  const float* W1c  = (const float*)d_in[12];  // (32,128)
  const float* b1c  = (const float*)d_in[13];
  const float* w2c  = (const float*)d_in[14];
  const float* b2c  = (const float*)d_in[15];
  const float* w3c  = (const float*)d_in[16];
  float* out = (float*)d_out;

  (void)hipFuncSetAttribute(reinterpret_cast<const void*>(cnf_logp_kernel),
                            hipFuncAttributeMaxDynamicSharedMemorySize, (int)SMEM_BYTES);
  cnf_logp_kernel<<<256, 128, SMEM_BYTES, stream>>>(
      z, cond, W1a, b1a, w2a, b2a, w3a,
      W1b, b1b, w2b, b2b, w3b,
      W1c, b1c, w2c, b2c, w3c, out);
}